// LGNNGINELayer_12463995093126
// MI455X (gfx1250) — compile-verified
//
#include <hip/hip_runtime.h>

#define D 128

typedef __attribute__((ext_vector_type(2))) float v2f;
typedef __attribute__((ext_vector_type(8))) float v8f;

__device__ __forceinline__ float relu1(float v) { return v > 0.0f ? v : 0.0f; }

// ---------------------------------------------------------------------------
// lgX[e] = 0.5 * (x[col0[e]] + x[col1[e]])
// ---------------------------------------------------------------------------
__global__ __launch_bounds__(256) void k_lgx_init(
    const float* __restrict__ x, const int* __restrict__ col0,
    const int* __restrict__ col1, float* __restrict__ lgX, int E) {
  long long tid = (long long)blockIdx.x * blockDim.x + threadIdx.x;
  long long total = (long long)E * (D / 4);
  if (tid >= total) return;
  int e = (int)(tid >> 5);
  int c = (int)(tid & 31) << 2;
  const float4 va = *(const float4*)(x + (long long)col0[e] * D + c);
  const float4 vb = *(const float4*)(x + (long long)col1[e] * D + c);
  float4 r;
  r.x = 0.5f * (va.x + vb.x);
  r.y = 0.5f * (va.y + vb.y);
  r.z = 0.5f * (va.z + vb.z);
  r.w = 0.5f * (va.w + vb.w);
  *(float4*)(lgX + (long long)e * D + c) = r;
}

// ---------------------------------------------------------------------------
// Pack weight matrix for WMMA B-fragment b64 loads:
// Wp[((kk*2+hi)*128 + n)*2 + {0,1}] = W[kk*4+hi*2 + {0,1}][n]
// ---------------------------------------------------------------------------
__global__ __launch_bounds__(256) void k_pack_w(
    const float* __restrict__ W, float* __restrict__ Wp) {
  int t = blockIdx.x * blockDim.x + threadIdx.x;  // (kk, hi, n)
  if (t >= 32 * 2 * D) return;
  int n = t & (D - 1);
  int hi = (t >> 7) & 1;
  int kk = t >> 8;
  int kb = (kk << 2) + (hi << 1);
  v2f v;
  v.x = W[(long long)kb * D + n];
  v.y = W[(long long)(kb + 1) * D + n];
  *(v2f*)(Wp + (long long)t * 2) = v;
}

// ---------------------------------------------------------------------------
// message + scatter-add: aggr[lg_dst[m]] += relu(lgX[lg_src[m]] + x[eai[m]])
// ---------------------------------------------------------------------------
__global__ __launch_bounds__(256) void k_message(
    const float* __restrict__ lgX, const float* __restrict__ x,
    const int* __restrict__ lg_src, const int* __restrict__ lg_dst,
    const int* __restrict__ eai, float* __restrict__ aggr, long long M) {
  long long tid = (long long)blockIdx.x * blockDim.x + threadIdx.x;
  long long total = M * (D / 4);
  if (tid >= total) return;
  long long m = tid >> 5;
  int c = (int)(tid & 31) << 2;
  long long s = lg_src[m];
  long long d = lg_dst[m];
  long long a = eai[m];
  const float4 vs = *(const float4*)(lgX + s * D + c);
  const float4 ve = *(const float4*)(x + a * D + c);
  float* dst = aggr + d * D + c;
  atomicAdd(dst + 0, relu1(vs.x + ve.x));
  atomicAdd(dst + 1, relu1(vs.y + ve.y));
  atomicAdd(dst + 2, relu1(vs.z + ve.z));
  atomicAdd(dst + 3, relu1(vs.w + ve.w));
}

// ---------------------------------------------------------------------------
// MLP: lgX = relu((lgX + aggr) @ W1 + b1) @ W2 + b2   (in-place over lgX rows)
// 4 waves per block, each wave owns a 16-row tile; fp32 WMMA 16x16x4.
// A 16x4 fragment: lanes 0-15 -> M=lane, {K, K+1}; lanes 16-31 -> {K+2, K+3}
// B 4x16 fragment: lanes 0-15 -> N=lane, rows {K, K+1}; lanes 16-31 -> {K+2, K+3}
// D layout: vgpr r -> M = r + 8*(lane>=16), N = lane&15
// B comes from packed Wp (one b64 per fragment).
// ---------------------------------------------------------------------------
__global__ __launch_bounds__(128) void k_mlp(
    const float* __restrict__ lgX, const float* __restrict__ aggr,
    const float* __restrict__ W1p, const float* __restrict__ b1,
    const float* __restrict__ W2p, const float* __restrict__ b2,
    float* __restrict__ outX, int E) {
  __shared__ float sMid[4][16 * 132];  // padded stride to spread LDS banks

  const int wave = threadIdx.x >> 5;
  const int lane = threadIdx.x & 31;
  const int arow = lane & 15;   // M (A) / N (B) within tile
  const int hi   = lane >> 4;   // selects K pair {0,1} vs {2,3}
  const int r0   = blockIdx.x * 64 + wave * 16;

  int grow = r0 + arow;
  if (grow > E - 1) grow = E - 1;  // clamp tail loads; stores guarded below
  const float* rowX = lgX + (long long)grow * D;
  const float* rowA = aggr + (long long)grow * D;

  const v8f vzero = {0.f, 0.f, 0.f, 0.f, 0.f, 0.f, 0.f, 0.f};
  v8f acc[8];
#pragma unroll
  for (int i = 0; i < 8; ++i) acc[i] = vzero;

  // ---- GEMM1: (lgX + aggr) @ W1 ----
  for (int kk = 0; kk < 32; ++kk) {
    const int kb = (kk << 2) + (hi << 1);
    const v2f ax = *(const v2f*)(rowX + kb);
    const v2f aa = *(const v2f*)(rowA + kb);
    const v2f a = ax + aa;
    const float* wp = W1p + ((((kk << 1) + hi) << 7) << 1);  // (kk,hi) group of 128 pairs
#pragma unroll
    for (int nt = 0; nt < 8; ++nt) {
      const int n = (nt << 4) + arow;
      const v2f b = *(const v2f*)(wp + (n << 1));
      acc[nt] = __builtin_amdgcn_wmma_f32_16x16x4_f32(
          false, a, false, b, (short)0, acc[nt], false, false);
    }
  }

  // bias + relu, relayout via LDS (D-layout -> A-layout for GEMM2)
#pragma unroll
  for (int nt = 0; nt < 8; ++nt) {
#pragma unroll
    for (int r = 0; r < 8; ++r) {
      const int row = r + (hi << 3);
      const int col = (nt << 4) + arow;
      sMid[wave][row * 132 + col] = relu1(acc[nt][r] + b1[col]);
    }
  }
  __syncthreads();

  v8f acc2[8];
#pragma unroll
  for (int i = 0; i < 8; ++i) acc2[i] = vzero;

  // ---- GEMM2: mid @ W2 ----
  for (int kk = 0; kk < 32; ++kk) {
    const int kb = (kk << 2) + (hi << 1);
    const v2f a = *(const v2f*)(&sMid[wave][arow * 132 + kb]);
    const float* wp = W2p + ((((kk << 1) + hi) << 7) << 1);
#pragma unroll
    for (int nt = 0; nt < 8; ++nt) {
      const int n = (nt << 4) + arow;
      const v2f b = *(const v2f*)(wp + (n << 1));
      acc2[nt] = __builtin_amdgcn_wmma_f32_16x16x4_f32(
          false, a, false, b, (short)0, acc2[nt], false, false);
    }
  }

  // bias + store (in-place into lgX; this block already consumed its rows)
#pragma unroll
  for (int nt = 0; nt < 8; ++nt) {
#pragma unroll
    for (int r = 0; r < 8; ++r) {
      const int row = r0 + r + (hi << 3);
      const int col = (nt << 4) + arow;
      if (row < E) outX[(long long)row * D + col] = acc2[nt][r] + b2[col];
    }
  }
}

// ---------------------------------------------------------------------------
// scatter-sum by col1: sums[col1[e]] += lgX[e]; cnt[col1[e]] += 1
// ---------------------------------------------------------------------------
__global__ __launch_bounds__(256) void k_scatter_out(
    const float* __restrict__ lgX, const int* __restrict__ col1,
    float* __restrict__ sums, float* __restrict__ cnt, int E) {
  long long tid = (long long)blockIdx.x * blockDim.x + threadIdx.x;
  long long total = (long long)E * (D / 4);
  if (tid >= total) return;
  int e = (int)(tid >> 5);
  int c = (int)(tid & 31) << 2;
  long long n = col1[e];
  const float4 v = *(const float4*)(lgX + (long long)e * D + c);
  float* dst = sums + n * D + c;
  atomicAdd(dst + 0, v.x);
  atomicAdd(dst + 1, v.y);
  atomicAdd(dst + 2, v.z);
  atomicAdd(dst + 3, v.w);
  if (c == 0) atomicAdd(cnt + n, 1.0f);
}

// ---------------------------------------------------------------------------
// out = x + relu(cnt > 0 ? sums / cnt : 0)
// ---------------------------------------------------------------------------
__global__ __launch_bounds__(256) void k_finalize(
    const float* __restrict__ x, const float* __restrict__ sums,
    const float* __restrict__ cnt, float* __restrict__ out, int N) {
  long long tid = (long long)blockIdx.x * blockDim.x + threadIdx.x;
  long long total = (long long)N * (D / 4);
  if (tid >= total) return;
  int n = (int)(tid >> 5);
  int c = (int)(tid & 31) << 2;
  const float cc = cnt[n];
  const float inv = cc > 0.0f ? (1.0f / cc) : 0.0f;
  const float4 s = *(const float4*)(sums + (long long)n * D + c);
  const float4 xv = *(const float4*)(x + (long long)n * D + c);
  float4 r;
  r.x = xv.x + relu1(s.x * inv);
  r.y = xv.y + relu1(s.y * inv);
  r.z = xv.z + relu1(s.z * inv);
  r.w = xv.w + relu1(s.w * inv);
  *(float4*)(out + (long long)n * D + c) = r;
}

// ---------------------------------------------------------------------------
extern "C" void kernel_launch(void* const* d_in, const int* in_sizes, int n_in,
                              void* d_out, int out_size, void* d_ws, size_t ws_size,
                              hipStream_t stream) {
  const float* x   = (const float*)d_in[0];
  const float* W1  = (const float*)d_in[1];
  const float* b1  = (const float*)d_in[2];
  const float* W2  = (const float*)d_in[3];
  const float* b2  = (const float*)d_in[4];
  const int* col0  = (const int*)d_in[5];
  const int* col1  = (const int*)d_in[6];
  const int* lgsrc = (const int*)d_in[7];
  const int* lgdst = (const int*)d_in[8];
  const int* eai   = (const int*)d_in[9];

  const int N = in_sizes[0] / D;
  const int E = in_sizes[5];
  const long long M = in_sizes[7];

  char* ws = (char*)d_ws;
  float* lgX  = (float*)ws;  ws += (size_t)E * D * sizeof(float);
  float* aggr = (float*)ws;  ws += (size_t)E * D * sizeof(float);
  float* sums = (float*)ws;  ws += (size_t)N * D * sizeof(float);
  float* cnt  = (float*)ws;  ws += (size_t)N * sizeof(float);
  float* W1p  = (float*)ws;  ws += (size_t)D * D * sizeof(float);
  float* W2p  = (float*)ws;
  (void)ws_size; (void)n_in; (void)out_size;

  // pack weights for b64 B-fragment loads (deterministic, every call)
  k_pack_w<<<(32 * 2 * D + 255) / 256, 256, 0, stream>>>(W1, W1p);
  k_pack_w<<<(32 * 2 * D + 255) / 256, 256, 0, stream>>>(W2, W2p);

  // lgX init
  {
    long long total = (long long)E * (D / 4);
    unsigned blocks = (unsigned)((total + 255) / 256);
    k_lgx_init<<<blocks, 256, 0, stream>>>(x, col0, col1, lgX, E);
  }

  for (int it = 0; it < 2; ++it) {
    hipMemsetAsync(aggr, 0, (size_t)E * D * sizeof(float), stream);
    if (M > 0) {
      long long total = M * (D / 4);
      unsigned blocks = (unsigned)((total + 255) / 256);
      k_message<<<blocks, 256, 0, stream>>>(lgX, x, lgsrc, lgdst, eai, aggr, M);
    }
    {
      unsigned blocks = (unsigned)((E + 63) / 64);
      k_mlp<<<blocks, 128, 0, stream>>>(lgX, aggr, W1p, b1, W2p, b2, lgX, E);
    }
  }

  hipMemsetAsync(sums, 0, (size_t)N * D * sizeof(float), stream);
  hipMemsetAsync(cnt, 0, (size_t)N * sizeof(float), stream);
  {
    long long total = (long long)E * (D / 4);
    unsigned blocks = (unsigned)((total + 255) / 256);
    k_scatter_out<<<blocks, 256, 0, stream>>>(lgX, col1, sums, cnt, E);
  }
  {
    long long total = (long long)N * (D / 4);
    unsigned blocks = (unsigned)((total + 255) / 256);
    k_finalize<<<blocks, 256, 0, stream>>>(x, sums, cnt, (float*)d_out, N);
  }
}